// ImproveFFBlock4_83597243450173
// MI455X (gfx1250) — compile-verified
//
#include <hip/hip_runtime.h>
#include <stdint.h>

typedef float vfloat2 __attribute__((ext_vector_type(2)));
typedef float vfloat4 __attribute__((ext_vector_type(4)));

#define NEG_INF (-3.402823466e+38f)

__device__ __forceinline__ float max4(vfloat4 v) {
    return fmaxf(fmaxf(v.x, v.y), fmaxf(v.z, v.w));
}

// All tensors are <= 268 MB, so byte offsets fit in u32: index from the
// uniform base with an unsigned byte offset so the backend emits the
// SADDR + 32-bit-voffset global load form (cheap scalar base, 1 VALU addr).
__device__ __forceinline__ vfloat2 ldnt2(const float* base, unsigned boff) {
    return __builtin_nontemporal_load((const vfloat2*)((const char*)base + boff));
}
__device__ __forceinline__ vfloat4 ldnt4(const float* base, unsigned boff) {
    return __builtin_nontemporal_load((const vfloat4*)((const char*)base + boff));
}

// ---------------------------------------------------------------------------
// pool2: in1 (8,256,64,64) -> p1 (8,256,32,32). 1 thread / output, float2
// loads; consecutive lanes read consecutive 8B -> fully coalesced.
// ---------------------------------------------------------------------------
__global__ void pool2_k(const float* __restrict__ in, float* __restrict__ out) {
    unsigned tid = blockIdx.x * blockDim.x + threadIdx.x;   // 2,097,152 outputs
    unsigned plane = tid >> 10;
    unsigned h = (tid >> 5) & 31, w = tid & 31;
    unsigned boff = (plane << 14) + (h << 9) + (w << 3);    // bytes
    vfloat2 a = ldnt2(in, boff);
    vfloat2 b = ldnt2(in, boff + 256);                      // next input row
    out[tid] = fmaxf(fmaxf(a.x, a.y), fmaxf(b.x, b.y));
}

// ---------------------------------------------------------------------------
// pool4: in2 (8,128,128,128) -> p2 (8,128,32,32). 1 thread / output; each row
// of the 4x4 window is one b128 load; lanes stride exactly 16B -> coalesced.
// ---------------------------------------------------------------------------
__global__ void pool4_k(const float* __restrict__ in, float* __restrict__ out) {
    unsigned tid = blockIdx.x * blockDim.x + threadIdx.x;   // 1,048,576 outputs
    unsigned plane = tid >> 10;
    unsigned h = (tid >> 5) & 31, w = tid & 31;
    unsigned boff = (plane << 16) + (h << 11) + (w << 4);   // bytes
    float m = NEG_INF;
#pragma unroll
    for (int r = 0; r < 4; ++r)
        m = fmaxf(m, max4(ldnt4(in, boff + r * 512)));      // +128 floats/row
    out[tid] = m;
}

// ---------------------------------------------------------------------------
// pool8: in3 (8,64,256,256) -> p3 (8,64,32,32). 2 lanes cooperate per output:
// lane j covers columns 8w+4j..+3 over 8 rows (b128 each); byte addresses
// across the wave are contiguous. Cross-lane max via shfl_xor (wave32).
// ---------------------------------------------------------------------------
__global__ void pool8_k(const float* __restrict__ in, float* __restrict__ out) {
    unsigned tid = blockIdx.x * blockDim.x + threadIdx.x;   // 2 * 524,288 threads
    unsigned o = tid >> 1, j = tid & 1;
    unsigned plane = o >> 10;
    unsigned h = (o >> 5) & 31, w = o & 31;
    unsigned boff = (plane << 18) + (h << 13) + (w << 5) + (j << 4);  // bytes
    float m = NEG_INF;
#pragma unroll
    for (int r = 0; r < 8; ++r)
        m = fmaxf(m, max4(ldnt4(in, boff + r * 1024)));     // +256 floats/row
    m = fmaxf(m, __shfl_xor(m, 1, 32));
    if (j == 0) out[o] = m;
}

// ---------------------------------------------------------------------------
// pool16: in4 (8,32,512,512) -> p4 (8,32,32,32). Dominant stream (268 MB).
// One workgroup per (plane, out-row): the 16 input rows are one contiguous
// 32KB block. Stage it into LDS with the CDNA5 async-copy path
// (global_load_async_to_lds_b128, tracked by ASYNCcnt), s_wait_asynccnt +
// barrier, then 8 lanes per output reduce the 16x16 window from LDS
// (ds_load_b128) and combine with shfl_xor. blockDim = 256.
// ---------------------------------------------------------------------------
__global__ void pool16_async_k(const float* __restrict__ in,
                               float* __restrict__ out) {
    __shared__ __align__(16) float smem[8192];              // 32 KB band
    const int tid = threadIdx.x;
    const unsigned plane = blockIdx.x >> 5;
    const unsigned h = blockIdx.x & 31;

    // Uniform 64-bit base of this block's contiguous 32KB source region.
    const unsigned long long gbase =
        (unsigned long long)(uintptr_t)(in + ((size_t)plane << 18) + ((size_t)h << 13));
    // LDS byte offset of smem (low 32 bits of the flat shared address).
    const unsigned lds_base = (unsigned)(uintptr_t)(void*)smem;

    // 256 threads x 8 iters x 16B = 32KB, byte-contiguous per iteration.
#pragma unroll
    for (int i = 0; i < 8; ++i) {
        unsigned off = (unsigned)(i * 4096 + tid * 16);
        unsigned lds = lds_base + off;
        asm volatile("global_load_async_to_lds_b128 %0, %1, %2"
                     :
                     : "v"(lds), "v"(off), "s"(gbase)
                     : "memory");
    }
    asm volatile("s_wait_asynccnt 0" ::: "memory");
    __syncthreads();

    // 8 lanes per output column: lane j reduces rows 2j,2j+1 of the window.
    const int j = tid & 7;
    const int w = tid >> 3;                                 // 0..31
    float m = NEG_INF;
#pragma unroll
    for (int r2 = 0; r2 < 2; ++r2) {
        const int row = j * 2 + r2;
        const vfloat4* sp = (const vfloat4*)(smem + row * 512 + w * 16);
#pragma unroll
        for (int c = 0; c < 4; ++c) m = fmaxf(m, max4(sp[c]));
    }
    m = fmaxf(m, __shfl_xor(m, 1, 32));
    m = fmaxf(m, __shfl_xor(m, 2, 32));
    m = fmaxf(m, __shfl_xor(m, 4, 32));
    if (j == 0) out[(blockIdx.x << 5) + w] = m;             // plane*1024 + h*32 + w
}

// ---------------------------------------------------------------------------
// combine: out[b,c,h,w] = relu(p1[b,c%256,hw] + p2[b,c%128,hw] + p3[b,c%64,hw]
//                              + p4[b,c%32,hw] + ff[b,c,hw]).
// float4 per thread; p re-reads hit L2 (15.7 MB resident vs 192 MB L2); the
// pooled-array loads stay temporal, the ff stream + final store are NT.
// ---------------------------------------------------------------------------
__global__ void combine_k(const float* __restrict__ p1, const float* __restrict__ p2,
                          const float* __restrict__ p3, const float* __restrict__ p4,
                          const float* __restrict__ ff, float* __restrict__ out) {
    unsigned t = blockIdx.x * blockDim.x + threadIdx.x;     // 1,048,576 float4s
    unsigned e = t << 2;                                    // element index
    unsigned hwb = (e & 1023) << 2;                         // hw byte offset
    unsigned c = (e >> 10) & 511;
    unsigned b = e >> 19;
    vfloat4 v1 = *(const vfloat4*)((const char*)p1 + ((((b << 8) + (c & 255)) << 12) + hwb));
    vfloat4 v2 = *(const vfloat4*)((const char*)p2 + ((((b << 7) + (c & 127)) << 12) + hwb));
    vfloat4 v3 = *(const vfloat4*)((const char*)p3 + ((((b << 6) + (c &  63)) << 12) + hwb));
    vfloat4 v4 = *(const vfloat4*)((const char*)p4 + ((((b << 5) + (c &  31)) << 12) + hwb));
    vfloat4 vf = ldnt4(ff, e << 2);
    vfloat4 s = v1 + v2 + v3 + v4 + vf;
    s.x = fmaxf(s.x, 0.0f);
    s.y = fmaxf(s.y, 0.0f);
    s.z = fmaxf(s.z, 0.0f);
    s.w = fmaxf(s.w, 0.0f);
    __builtin_nontemporal_store(s, (vfloat4*)((char*)out + (e << 2)));
}

extern "C" void kernel_launch(void* const* d_in, const int* in_sizes, int n_in,
                              void* d_out, int out_size, void* d_ws, size_t ws_size,
                              hipStream_t stream) {
    (void)in_sizes; (void)n_in; (void)out_size; (void)ws_size;
    const float* in1 = (const float*)d_in[0];  // (8,256, 64, 64)
    const float* in2 = (const float*)d_in[1];  // (8,128,128,128)
    const float* in3 = (const float*)d_in[2];  // (8, 64,256,256)
    const float* in4 = (const float*)d_in[3];  // (8, 32,512,512)
    const float* ff  = (const float*)d_in[4];  // (8,512, 32, 32)
    float* out = (float*)d_out;

    // Workspace partition (floats): 15.7 MB total.
    float* p1 = (float*)d_ws;          // 2,097,152
    float* p2 = p1 + 2097152;          // 1,048,576
    float* p3 = p2 + 1048576;          //   524,288
    float* p4 = p3 + 524288;           //   262,144

    pool2_k       <<<8192, 256, 0, stream>>>(in1, p1);
    pool4_k       <<<4096, 256, 0, stream>>>(in2, p2);
    pool8_k       <<<4096, 256, 0, stream>>>(in3, p3);
    pool16_async_k<<<8192, 256, 0, stream>>>(in4, p4);
    combine_k     <<<4096, 256, 0, stream>>>(p1, p2, p3, p4, ff, out);
}